// SGFE_13417477833522
// MI455X (gfx1250) — compile-verified
//
#include <hip/hip_runtime.h>
#include <hip/hip_bf16.h>
#include <math.h>

typedef float v2f __attribute__((ext_vector_type(2)));
typedef float v8f __attribute__((ext_vector_type(8)));

// ---------------- problem constants ----------------
constexpr int NPTS = 200000;
constexpr int CIN  = 64;
constexpr int CR   = 64;
constexpr int CR2  = 32;
constexpr int GX2c = 240, GY2c = 180, GZ2c = 16;

// per-scale segment grids: ps = {2,4,6,8}
constexpr int PSV[4]  = {2, 4, 6, 8};
constexpr int DXV[4]  = {240, 120, 80, 60};
constexpr int DYV[4]  = {180,  90, 60, 45};
constexpr int DZV[4]  = { 16,   8,  6,  4};
constexpr int NSEGV[4]  = {1382400, 172800, 57600, 21600};
constexpr int SEGOFF[4] = {0, 1382400, 1555200, 1612800};
constexpr int SEGTOT    = 1634400;

constexpr long long PROJ_ELEMS = (long long)NPTS * CR;              // 12,800,000
constexpr long long TV_SEGS    = 2LL * GZ2c * GY2c * GX2c;          // 1,382,400
constexpr long long TV_ELEMS   = TV_SEGS * CR;                      // 88,473,600

// ---------------- WMMA multi-tile helper ----------------
// One wave computes NT adjacent 16x16 f32 tiles sharing the same A rows.
// A_lane: pointer to THIS lane's A row (row M = lane&15); ascale multiplies A.
// B: row-major K x Ncols, leading dim ldb; tiles at col0, col0+16, ...
// Uniform kk loop (lane-varying k0 only in the address) -> no exec masking,
// fully unrolled so loads batch into clauses ahead of the WMMA chain.
template <int K, int NT>
__device__ __forceinline__ void wmma_mt(const float* __restrict__ A_lane,
                                        float ascale,
                                        const float* __restrict__ B, int ldb,
                                        int col0, int lane, v8f* acc) {
  const int m     = lane & 15;
  const int khalf = (lane < 16) ? 0 : 2;
#pragma unroll
  for (int kk = 0; kk < K / 4; ++kk) {
    const int k0 = kk * 4 + khalf;
    v2f a;
    a.x = A_lane[k0] * ascale;
    a.y = A_lane[k0 + 1] * ascale;
#pragma unroll
    for (int t = 0; t < NT; ++t) {
      v2f b;
      b.x = B[(size_t)k0 * ldb + col0 + t * 16 + m];
      b.y = B[(size_t)(k0 + 1) * ldb + col0 + t * 16 + m];
      acc[t] = __builtin_amdgcn_wmma_f32_16x16x4_f32(false, a, false, b,
                                                     (short)0, acc[t],
                                                     false, false);
    }
  }
}

// ---------------- K2: reduced = relu(X @ W_red + b_red) ----------------
// 256 threads = 8 waves; wave: 16 rows x 32 cols; block: 64 rows x 64 cols.
__global__ void k_reduce(const float* __restrict__ X, const float* __restrict__ W,
                         const float* __restrict__ bias, float* __restrict__ R) {
  const int lane = threadIdx.x & 31;
  const int w    = threadIdx.x >> 5;
  const int row0 = blockIdx.x * 64 + (w >> 1) * 16;
  const int col0 = (w & 1) * 32;
  const int m    = lane & 15;

  v8f acc[2] = {};
  wmma_mt<64, 2>(X + (size_t)(row0 + m) * CIN, 1.0f, W, CR, col0, lane, acc);

  const int rb = (lane < 16) ? 0 : 8;
#pragma unroll
  for (int t = 0; t < 2; ++t) {
    const int   col = col0 + t * 16 + m;
    const float bv  = bias[col];
#pragma unroll
    for (int r = 0; r < 8; ++r) {
      const float v = acc[t][r] + bv;
      R[(size_t)(row0 + rb + r) * CR + col] = v > 0.f ? v : 0.f;
    }
  }
}

// ---------------- K_lin: per-point segment ids for the 4 scales ----------------
__global__ void k_lin(const int* __restrict__ coords, int* __restrict__ lin) {
  const int i = blockIdx.x * blockDim.x + threadIdx.x;
  if (i >= NPTS) return;
  const int b = coords[i * 4 + 0];
  const int x = coords[i * 4 + 1];
  const int y = coords[i * 4 + 2];
  const int z = coords[i * 4 + 3];
#pragma unroll
  for (int j = 0; j < 4; ++j) {
    const int qx = x / PSV[j], qy = y / PSV[j], qz = z / PSV[j];
    lin[j * NPTS + i] = ((b * DXV[j] + qx) * DYV[j] + qy) * DZV[j] + qz;
  }
}

// ---------------- K_scatter: segment sums + counts (atomic) ----------------
__global__ void k_scatter(const float* __restrict__ R, const int* __restrict__ lin,
                          float* __restrict__ sums, float* __restrict__ cnts) {
  const long long t = (long long)blockIdx.x * blockDim.x + threadIdx.x;
  if (t >= (long long)NPTS * CR) return;
  const int i = (int)(t >> 6);
  const int c = (int)(t & 63);
  const float v = R[t];
#pragma unroll
  for (int j = 0; j < 4; ++j) {
    const int s = lin[j * NPTS + i];
    atomicAdd(&sums[(size_t)(SEGOFF[j] + s) * 64 + c], v);
    if (c == 0) atomicAdd(&cnts[SEGOFF[j] + s], 1.0f);
  }
}

// ---------------- K_att: att = relu(mean_seg @ fc_list_W[j] + b), in place ----------------
// 256 threads = 8 waves; wave handles 16 segments, cols 0..31 (two tiles, shared A).
__global__ void k_att(float* __restrict__ sums, const float* __restrict__ cnts,
                      const float* __restrict__ W, const float* __restrict__ bias,
                      int nseg) {
  const int lane = threadIdx.x & 31;
  const int w    = threadIdx.x >> 5;
  const int seg0 = (blockIdx.x * 8 + w) * 16;
  if (seg0 >= nseg) return;                       // wave-uniform branch

  const int m = lane & 15;
  int row = seg0 + m;
  if (row >= nseg) row = nseg - 1;                // clamp loads; EXEC stays full
  const float ic = 1.0f / fmaxf(cnts[row], 1.0f);

  v8f acc[2] = {};
  wmma_mt<64, 2>(sums + (size_t)row * 64, ic, W, CR2, 0, lane, acc);

  const int   rb = (lane < 16) ? 0 : 8;
#pragma unroll
  for (int t = 0; t < 2; ++t) {
    const float bv = bias[t * 16 + m];
#pragma unroll
    for (int r = 0; r < 8; ++r) {
      const int rr = seg0 + rb + r;
      if (rr < nseg) {
        const float v = acc[t][r] + bv;
        sums[(size_t)rr * 64 + t * 16 + m] = v > 0.f ? v : 0.f;
      }
    }
  }
}

// ---------------- K_fuse: per-point attention fusion + output MLP ----------------
// 64 threads = 2 waves; each wave processes 16 points through LDS staging.
// per-wave LDS: SF[4*16*32] + Abuf[16*128] + Dbuf[16*64] = 5120 floats (20KB)
__global__ void k_fuse(const float* __restrict__ sums, const int* __restrict__ lin,
                       const float* __restrict__ R,
                       const float* __restrict__ fc_W,     // 32x32
                       const float* __restrict__ fcs_W,    // 4x32x32
                       const float* __restrict__ fcs_b,    // 4x32
                       const float* __restrict__ out_fc_W, // 32x64
                       const float* __restrict__ lo_W1,    // 128x64
                       const float* __restrict__ lo_W2,    // 64x64
                       const float* __restrict__ lo_b2,    // 64
                       float* __restrict__ proj_pre) {
  __shared__ float smem[2 * 5120];
  const int lane = threadIdx.x & 31;
  const int w    = threadIdx.x >> 5;
  float* SF   = smem + w * 5120;   // [4][16][32]
  float* Abuf = SF + 2048;         // [16][128]
  float* Dbuf = Abuf + 2048;       // [16][64]

  const int p0 = (blockIdx.x * 2 + w) * 16;   // NPTS % 32 == 0 -> no edges
  const int m  = lane & 15;
  const int rb = (lane < 16) ? 0 : 8;

  // prefetch the reduced rows consumed late (stage F)
  if (lane < 16) __builtin_prefetch(R + (size_t)(p0 + lane) * CR, 0, 1);

  // A) gather scale features + feat_S
  for (int t = lane; t < 512; t += 32) {
    const int pt = t >> 5, c = t & 31;
    const int gp = p0 + pt;
    float s = 0.f;
#pragma unroll
    for (int k = 0; k < 4; ++k) {
      const int seg = lin[k * NPTS + gp];
      const float v = sums[(size_t)(SEGOFF[k] + seg) * 64 + c];
      SF[(k * 16 + pt) * 32 + c] = v;
      s += v;
    }
    Abuf[pt * 128 + c] = s;
  }
  __syncthreads();

  // B) feat_Z = relu(feat_S @ fc_W)   (16x32 @ 32x32)
  {
    v8f acc[2] = {};
    wmma_mt<32, 2>(Abuf + m * 128, 1.0f, fc_W, CR2, 0, lane, acc);
#pragma unroll
    for (int t = 0; t < 2; ++t) {
      const int col = t * 16 + m;
#pragma unroll
      for (int r = 0; r < 8; ++r) {
        const float v = acc[t][r];
        Dbuf[(rb + r) * 64 + col] = v > 0.f ? v : 0.f;
      }
    }
  }
  __syncthreads();

  // C) att_v[k] = sigmoid(feat_Z @ fcs_W[k] + fcs_b[k]); SF *= att_v
#pragma unroll
  for (int k = 0; k < 4; ++k) {
    v8f acc[2] = {};
    wmma_mt<32, 2>(Dbuf + m * 64, 1.0f, fcs_W + k * 32 * 32, CR2, 0, lane, acc);
#pragma unroll
    for (int t = 0; t < 2; ++t) {
      const int   col = t * 16 + m;
      const float bv  = fcs_b[k * 32 + col];
#pragma unroll
      for (int r = 0; r < 8; ++r) {
        const float sg = 1.0f / (1.0f + __expf(-(acc[t][r] + bv)));
        SF[(k * 16 + rb + r) * 32 + col] *= sg;
      }
    }
  }
  __syncthreads();

  // D) fused_pre = sum_k SF[k]  -> Abuf cols 0..31
  for (int t = lane; t < 512; t += 32) {
    const int pt = t >> 5, c = t & 31;
    Abuf[pt * 128 + c] = SF[pt * 32 + c] + SF[(16 + pt) * 32 + c] +
                         SF[(32 + pt) * 32 + c] + SF[(48 + pt) * 32 + c];
  }
  __syncthreads();

  // E) fused = fused_pre @ out_fc_W (16x32 @ 32x64) -> Abuf cols 64..127
  {
    v8f acc[4] = {};
    wmma_mt<32, 4>(Abuf + m * 128, 1.0f, out_fc_W, CR, 0, lane, acc);
#pragma unroll
    for (int t = 0; t < 4; ++t) {
      const int col = t * 16 + m;
#pragma unroll
      for (int r = 0; r < 8; ++r) Abuf[(rb + r) * 128 + 64 + col] = acc[t][r];
    }
  }
  __syncthreads();

  // F) output_f cols 0..63 = reduced rows
  for (int t = lane; t < 16 * 64; t += 32) {
    const int pt = t >> 6, c = t & 63;
    Abuf[pt * 128 + c] = R[(size_t)(p0 + pt) * CR + c];
  }
  __syncthreads();

  // G) h = relu(output_f @ lo_W1)  (16x128 @ 128x64) -> Dbuf
  {
    v8f acc[4] = {};
    wmma_mt<128, 4>(Abuf + m * 128, 1.0f, lo_W1, CR, 0, lane, acc);
#pragma unroll
    for (int t = 0; t < 4; ++t) {
      const int col = t * 16 + m;
#pragma unroll
      for (int r = 0; r < 8; ++r) {
        const float v = acc[t][r];
        Dbuf[(rb + r) * 64 + col] = v > 0.f ? v : 0.f;
      }
    }
  }
  __syncthreads();

  // H) proj = h @ lo_W2 + lo_b2 -> global
  {
    v8f acc[4] = {};
    wmma_mt<64, 4>(Dbuf + m * 64, 1.0f, lo_W2, CR, 0, lane, acc);
#pragma unroll
    for (int t = 0; t < 4; ++t) {
      const int   col = t * 16 + m;
      const float bv  = lo_b2[col];
#pragma unroll
      for (int r = 0; r < 8; ++r)
        proj_pre[(size_t)(p0 + rb + r) * CR + col] = acc[t][r] + bv;
    }
  }
}

// ---------------- monotonic float<->uint order map for segment max ----------------
__device__ __forceinline__ unsigned f2key(float f) {
  unsigned u = __float_as_uint(f);
  return (u & 0x80000000u) ? ~u : (u | 0x80000000u);
}
__device__ __forceinline__ float key2f(unsigned k) {
  unsigned u = (k & 0x80000000u) ? (k & 0x7FFFFFFFu) : ~k;
  return __uint_as_float(u);
}

// ---------------- K_gather_max: proj permutation gather + lin2 + seg max ----------------
__global__ void k_gather_max(const float* __restrict__ proj_pre,
                             const int* __restrict__ inv,
                             const int* __restrict__ bxyz,
                             float* __restrict__ out_proj,
                             unsigned* __restrict__ tvkeys,
                             int* __restrict__ out_lin2) {
  const long long t = (long long)blockIdx.x * blockDim.x + threadIdx.x;
  if (t >= (long long)NPTS * CR) return;
  const int i = (int)(t >> 6);
  const int c = (int)(t & 63);

  const float v = proj_pre[(size_t)inv[i] * CR + c];
  out_proj[t] = v;

  const int b = bxyz[i * 4 + 0];
  const int x = bxyz[i * 4 + 1];
  const int y = bxyz[i * 4 + 2];
  const int z = bxyz[i * 4 + 3];
  const int l2 = ((b * GZ2c + z) * GY2c + y) * GX2c + x;
  if (c == 0) out_lin2[i] = l2;

  atomicMax(&tvkeys[(size_t)l2 * CR + c], f2key(v));
}

// ---------------- K_decode: keys -> floats in place (empty seg -> 0) ----------------
__global__ void k_decode(unsigned* __restrict__ tv) {
  const long long t = (long long)blockIdx.x * blockDim.x + threadIdx.x;
  if (t >= TV_ELEMS) return;
  const unsigned k = tv[t];
  ((float*)tv)[t] = (k == 0u) ? 0.0f : key2f(k);
}

// ---------------- host launch ----------------
extern "C" void kernel_launch(void* const* d_in, const int* in_sizes, int n_in,
                              void* d_out, int out_size, void* d_ws, size_t ws_size,
                              hipStream_t stream) {
  const float* input_data = (const float*)d_in[0];
  const int*   coords     = (const int*)d_in[1];
  const int*   inv        = (const int*)d_in[2];
  const int*   bxyz       = (const int*)d_in[3];
  const float* W_red      = (const float*)d_in[4];
  const float* b_red      = (const float*)d_in[5];
  const float* fc_list_W  = (const float*)d_in[6];
  const float* fc_list_b  = (const float*)d_in[7];
  const float* fcs_W      = (const float*)d_in[8];
  const float* fcs_b      = (const float*)d_in[9];
  const float* fc_W       = (const float*)d_in[10];
  const float* out_fc_W   = (const float*)d_in[11];
  const float* lo_W1      = (const float*)d_in[12];
  const float* lo_W2      = (const float*)d_in[13];
  const float* lo_b2      = (const float*)d_in[14];

  // workspace layout (bytes)
  char* ws = (char*)d_ws;
  float* reduced  = (float*)(ws);                                      // N*64
  float* proj_pre = (float*)(ws + 51200000);                           // N*64
  int*   lin      = (int*)  (ws + 102400000);                          // 4*N
  float* cnts     = (float*)(ws + 105600000);                          // SEGTOT
  float* sums     = (float*)(ws + 105600000 + (size_t)SEGTOT * 4);     // SEGTOT*64

  // output layout
  float*    out_proj = (float*)d_out;
  unsigned* tvkeys   = (unsigned*)((char*)d_out + PROJ_ELEMS * 4);
  int*      out_lin2 = (int*)((char*)d_out + (PROJ_ELEMS + TV_ELEMS) * 4);

  // zero accumulators: counts + sums (contiguous), tv keys
  hipMemsetAsync(cnts, 0, (size_t)SEGTOT * 4 + (size_t)SEGTOT * 64 * 4, stream);
  hipMemsetAsync(tvkeys, 0, (size_t)TV_ELEMS * 4, stream);

  // 1) reduced = relu(X @ W_red + b_red)
  k_reduce<<<NPTS / 64, 256, 0, stream>>>(input_data, W_red, b_red, reduced);

  // 2) segment ids
  k_lin<<<(NPTS + 255) / 256, 256, 0, stream>>>(coords, lin);

  // 3) scatter-add segment sums/counts
  {
    long long tot = (long long)NPTS * CR;
    k_scatter<<<(unsigned)((tot + 255) / 256), 256, 0, stream>>>(reduced, lin, sums, cnts);
  }

  // 4) per-scale segment attention GEMM (in-place over sums cols 0..31)
  for (int j = 0; j < 4; ++j) {
    const int nseg = NSEGV[j];
    float* sums_j = sums + (size_t)SEGOFF[j] * 64;
    const float* cnts_j = cnts + SEGOFF[j];
    k_att<<<(nseg + 127) / 128, 256, 0, stream>>>(
        sums_j, cnts_j, fc_list_W + j * 64 * 32, fc_list_b + j * 32, nseg);
  }

  // 5) per-point fusion + output MLP
  k_fuse<<<NPTS / 32, 64, 0, stream>>>(sums, lin, reduced, fc_W, fcs_W, fcs_b,
                                       out_fc_W, lo_W1, lo_W2, lo_b2, proj_pre);

  // 6) gather + lin2 + segment max
  {
    long long tot = (long long)NPTS * CR;
    k_gather_max<<<(unsigned)((tot + 255) / 256), 256, 0, stream>>>(
        proj_pre, inv, bxyz, out_proj, tvkeys, out_lin2);
  }

  // 7) decode tv keys to floats in place
  k_decode<<<(unsigned)((TV_ELEMS + 255) / 256), 256, 0, stream>>>(tvkeys);
}